// VocabParallelEmbeddingWithLoRA_65755949301993
// MI455X (gfx1250) — compile-verified
//
#include <hip/hip_runtime.h>

typedef __attribute__((ext_vector_type(2))) float v2f;
typedef __attribute__((ext_vector_type(8))) float v8f;

#define VOCAB 32000
#define EMBED 2048
#define NADP  8
#define RANK  16

// ---- pass 1: adapter histogram ------------------------------------------
__global__ void lora_hist_kernel(const int* __restrict__ tw,
                                 int* __restrict__ counts, int T) {
  int t = blockIdx.x * blockDim.x + threadIdx.x;
  if (t < T) atomicAdd(&counts[tw[t]], 1);
}

// ---- pass 2: 16-aligned segment starts (single thread, 8 elements) ------
__global__ void lora_prefix_kernel(const int* __restrict__ counts,
                                   int* __restrict__ seg,
                                   int* __restrict__ cursor) {
  if (threadIdx.x == 0) {
    int acc = 0;
    for (int l = 0; l < NADP; ++l) {
      seg[l] = acc;
      cursor[l] = acc;
      acc += (counts[l] + 15) & ~15;   // pad each segment to a 16-row tile
    }
    seg[NADP] = acc;
  }
}

// ---- pass 3: scatter tokens into padded segments + gather LoRA-A vectors -
__global__ void lora_scatter_kernel(const int* __restrict__ ids,
                                    const int* __restrict__ tw,
                                    const float* __restrict__ A,   // [L][R][V]
                                    int* __restrict__ cursor,
                                    int* __restrict__ perm,
                                    float* __restrict__ aperm,     // [PADT][R]
                                    int T) {
  int t = blockIdx.x * blockDim.x + threadIdx.x;
  if (t >= T) return;
  int l = tw[t];
  int id = ids[t];
  int p = atomicAdd(&cursor[l], 1);
  perm[p] = t;
  const float* Al = A + (size_t)l * RANK * VOCAB + id;
  float av[RANK];
#pragma unroll
  for (int r = 0; r < RANK; ++r)       // batch the 16 strided gathers
    av[r] = Al[(size_t)r * VOCAB];
#pragma unroll
  for (int r = 0; r < RANK; ++r)
    aperm[(size_t)p * RANK + r] = av[r];
}

// ---- pass 4: per-adapter GEMM via V_WMMA_F32_16X16X4_F32 + fused base ----
__global__ __launch_bounds__(256) void lora_wmma_gemm_kernel(
    const int* __restrict__ ids,
    const float* __restrict__ weight,   // [V][D]
    const float* __restrict__ B,        // [L][D][R]
    const int* __restrict__ seg,        // [L+1], 16-aligned
    const int* __restrict__ perm,       // [PADT], -1 = padding row
    const float* __restrict__ aperm,    // [PADT][R]
    float* __restrict__ out,            // [T][D]
    float* __restrict__ sink) {         // write-only junk target for pad rows
  const int row0 = blockIdx.x * 16;
  if (row0 >= seg[NADP]) return;        // block-uniform: EXEC stays all-ones

  const int wave   = threadIdx.x >> 5;  // 8 waves: 8 consecutive 16-col tiles
  const int lane   = threadIdx.x & 31;
  const int lane16 = lane & 15;
  const int hi     = lane >> 4;
  const int col0   = blockIdx.y * 128 + wave * 16;

  // Which adapter owns this row tile (segments are 16-aligned, so uniform).
  int l = 0;
#pragma unroll
  for (int i = 1; i < NADP; ++i) l += (row0 >= seg[i]) ? 1 : 0;

  const float* Arow = aperm + ((size_t)row0 + lane16) * RANK;                 // A[m][*]
  const float* Brow = B + (size_t)l * EMBED * RANK + (size_t)(col0 + lane16) * RANK; // B[l][n][*]

  // Pull the 8 perm entries for this half-wave as two aligned 16B loads.
  const int4 pa = *(const int4*)(perm + row0 + hi * 8);
  const int4 pb = *(const int4*)(perm + row0 + hi * 8 + 4);
  const int tok[8] = {pa.x, pa.y, pa.z, pa.w, pb.x, pb.y, pb.z, pb.w};

  int idv[8];
#pragma unroll
  for (int i = 0; i < 8; ++i)
    idv[i] = ids[tok[i] < 0 ? 0 : tok[i]];          // clamp: always in-bounds

  float basev[8];
#pragma unroll
  for (int i = 0; i < 8; ++i)
    basev[i] = weight[(size_t)idv[i] * EMBED + col0 + lane16];

  // fp32 16x16x4 operand layout: lanes 0-15 hold K = {4c, 4c+1},
  // lanes 16-31 hold K = {4c+2, 4c+3}  -> one aligned float2 per lane.
  v8f acc = {};
#pragma unroll
  for (int c = 0; c < 4; ++c) {
    const int k = c * 4 + hi * 2;
    v2f av = *(const v2f*)(Arow + k);
    v2f bv = *(const v2f*)(Brow + k);
    acc = __builtin_amdgcn_wmma_f32_16x16x4_f32(
        /*neg_a=*/false, av, /*neg_b=*/false, bv,
        /*c_mod=*/(short)0, acc, /*reuse_a=*/false, /*reuse_b=*/false);
  }

  // Branch-free epilogue: D[m][n] = acc[i], m = i + 8*hi, n = lane16.
  // Pad rows redirect to a junk sink so every store is unconditional
  // (no EXEC churn) and non-temporal (written once, never re-read).
  float* const sinkp = sink + lane;
#pragma unroll
  for (int i = 0; i < 8; ++i) {
    const size_t off = (size_t)(tok[i] < 0 ? 0 : tok[i]) * EMBED + col0 + lane16;
    float* dst = (tok[i] >= 0) ? (out + off) : sinkp;
    __builtin_nontemporal_store(basev[i] + acc[i], dst);
  }
}

extern "C" void kernel_launch(void* const* d_in, const int* in_sizes, int n_in,
                              void* d_out, int out_size, void* d_ws, size_t ws_size,
                              hipStream_t stream) {
  const int*   ids    = (const int*)d_in[0];
  const int*   tw     = (const int*)d_in[1];
  const float* weight = (const float*)d_in[2];
  const float* A      = (const float*)d_in[3];
  const float* Bm     = (const float*)d_in[4];
  float*       out    = (float*)d_out;

  const int T    = in_sizes[0];
  const int PADT = T + NADP * 16;       // worst-case per-adapter 16-padding

  // workspace layout (ints): [0..7] counts, [8..16] seg, [20..27] cursor,
  // [32..32+PADT) perm, then float a_perm[PADT][RANK], then 64-float sink
  int*   counts = (int*)d_ws;
  int*   seg    = counts + 8;
  int*   cursor = counts + 20;
  int*   perm   = counts + 32;
  float* aperm  = (float*)(perm + PADT);
  float* sink   = aperm + (size_t)PADT * RANK;

  hipMemsetAsync(counts, 0, 8 * sizeof(int), stream);
  hipMemsetAsync(perm, 0xFF, (size_t)PADT * sizeof(int), stream);        // -1
  hipMemsetAsync(aperm, 0, (size_t)PADT * RANK * sizeof(float), stream); // pad rows = 0

  lora_hist_kernel<<<(T + 255) / 256, 256, 0, stream>>>(tw, counts, T);
  lora_prefix_kernel<<<1, 32, 0, stream>>>(counts, seg, cursor);
  lora_scatter_kernel<<<(T + 255) / 256, 256, 0, stream>>>(ids, tw, A, cursor,
                                                           perm, aperm, T);

  dim3 grid(PADT / 16, EMBED / 128);    // 16-row tiles x (8 waves * 16 cols)
  lora_wmma_gemm_kernel<<<grid, 256, 0, stream>>>(ids, weight, Bm, seg, perm,
                                                  aperm, out, sink);
}